// PDEEval_43851616092521
// MI455X (gfx1250) — compile-verified
//
#include <hip/hip_runtime.h>
#include <math.h>

#ifndef __has_builtin
#define __has_builtin(x) 0
#endif

#define BATCH 32
#define CH 4
#define HGT 512
#define WID 512
#define PLANE (HGT * WID)      // 262144
#define STEPS 8
#define NBLK 64                // reduction chunks per plane (== stencil blocks/batch)
#define CHUNK (PLANE / NBLK)   // 4096 floats per chunk
#define SROWS 8                // rows produced per stencil block

typedef __attribute__((ext_vector_type(2))) float v2f;
typedef __attribute__((ext_vector_type(8))) float v8f;
typedef __attribute__((ext_vector_type(4))) int v4i;

#define AS1 __attribute__((address_space(1)))
#define AS3 __attribute__((address_space(3)))

// ---- async global->LDS helpers (CDNA5 path with safe fallback) -------------
__device__ __forceinline__ void lds_copy_b128(const float* g, float* l) {
#if __has_builtin(__builtin_amdgcn_global_load_async_to_lds_b128)
  __builtin_amdgcn_global_load_async_to_lds_b128((AS1 v4i*)g, (AS3 v4i*)l, 0, 0);
#else
  *(float4*)l = *(const float4*)g;
#endif
}

__device__ __forceinline__ void lds_copy_wait() {
#if __has_builtin(__builtin_amdgcn_global_load_async_to_lds_b128)
#if __has_builtin(__builtin_amdgcn_s_wait_asynccnt)
  __builtin_amdgcn_s_wait_asynccnt(0);
#else
  asm volatile("s_wait_asynccnt 0" ::: "memory");
#endif
#endif
  __syncthreads();
}

// ---- one-time: copy channels 1..3 to out, block-partial sums ---------------
__global__ __launch_bounds__(256) void copy_partial_ch123(const float* __restrict__ x,
                                                          float* __restrict__ out,
                                                          float* __restrict__ partials3) {
  const int blk = blockIdx.x;            // 0..NBLK-1
  const int c = blockIdx.y + 1;          // 1..3
  const int b = blockIdx.z;              // 0..31
  const long off = ((long)b * CH + c) * PLANE + (long)blk * CHUNK;
  const float4* src = (const float4*)(x + off);
  float4* dst = (float4*)(out + off);
  const int tid = threadIdx.x;
  float s = 0.f;
#pragma unroll
  for (int k = 0; k < 4; ++k) {
    float4 q = src[tid + k * 256];
    dst[tid + k * 256] = q;
    s += q.x + q.y + q.z + q.w;
  }
  __shared__ float red[256];
  red[tid] = s;
  __syncthreads();
  for (int o = 128; o > 0; o >>= 1) {
    if (tid < o) red[tid] += red[tid + o];
    __syncthreads();
  }
  if (tid == 0) partials3[((long)b * 3 + (c - 1)) * NBLK + blk] = red[0];
}

__global__ void finalize_ch123(const float* __restrict__ partials3,
                               float* __restrict__ chanMeans) {
  const int i = threadIdx.x;  // 0..95
  if (i >= BATCH * 3) return;
  const int b = i / 3;
  const int c = i % 3 + 1;
  float s = 0.f;
#pragma unroll
  for (int k = 0; k < NBLK; ++k) s += partials3[(long)i * NBLK + k];
  chanMeans[b * CH + c] = s * (1.0f / (float)PLANE);
}

// ---- step-0 channel-0 partial sums (later steps fuse this into stencil) ----
__global__ __launch_bounds__(256) void reduce_plane(const float* __restrict__ src, long bstride,
                                                    float* __restrict__ partials) {
  const int blk = blockIdx.x;
  const int b = blockIdx.y;
  const float4* base = (const float4*)(src + (long)b * bstride + (long)blk * CHUNK);
  const int tid = threadIdx.x;
  float s = 0.f;
#pragma unroll
  for (int k = 0; k < 4; ++k) {
    float4 q = base[tid + k * 256];
    s += q.x + q.y + q.z + q.w;
  }
  __shared__ float red[256];
  red[tid] = s;
  __syncthreads();
  for (int o = 128; o > 0; o >>= 1) {
    if (tid < o) red[tid] += red[tid + o];
    __syncthreads();
  }
  if (tid == 0) partials[(long)b * NBLK + blk] = red[0];
}

// ---- per-step head: finish mean, WMMA feats@W + b, sigmoid, mask -> D,rho --
__global__ __launch_bounds__(32) void logit_head(const float* __restrict__ partials,
                                                 const float* __restrict__ chanMeans,
                                                 const float* __restrict__ Wg,
                                                 const float* __restrict__ bg,
                                                 const int* __restrict__ tg, int step,
                                                 float* __restrict__ Dm,
                                                 float* __restrict__ rhom) {
  __shared__ float sf[BATCH][CH];   // feats
  __shared__ float zs[BATCH][2];    // logits
  const int lane = threadIdx.x;     // == batch index, one full wave32, EXEC all-1s
  float s = 0.f;
#pragma unroll
  for (int k = 0; k < NBLK; ++k) s += partials[(long)lane * NBLK + k];
  sf[lane][0] = s * (1.0f / (float)PLANE);
  sf[lane][1] = chanMeans[lane * CH + 1];
  sf[lane][2] = chanMeans[lane * CH + 2];
  sf[lane][3] = chanMeans[lane * CH + 3];
  __syncthreads();

  // A: 16x4 f32  (lanes 0-15: K=0,1 ; lanes 16-31: K=2,3), two M-tiles of 16 batches
  // B: 4x16 f32  (VGPR0: K=0|K=2 rows, VGPR1: K=1|K=3 rows), N>=2 padded with 0
  // C: bias broadcast along M (C[m][n] = b[n] for n<2 else 0)
  const int r = lane & 15;
  const int n = lane & 15;
  const int k0 = (lane >> 4) * 2;
  v2f bv;
  bv.x = (n < 2) ? Wg[k0 * 2 + n] : 0.f;
  bv.y = (n < 2) ? Wg[(k0 + 1) * 2 + n] : 0.f;
  const float cval = (n < 2) ? bg[n] : 0.f;
  v8f cm;
#pragma unroll
  for (int g = 0; g < 8; ++g) cm[g] = cval;

#pragma unroll
  for (int tile = 0; tile < 2; ++tile) {
    v2f av;
    av.x = sf[tile * 16 + r][k0];
    av.y = sf[tile * 16 + r][k0 + 1];
    v8f d = __builtin_amdgcn_wmma_f32_16x16x4_f32(false, av, false, bv,
                                                  (short)0, cm, false, false);
    if (n < 2) {
      const int mbase = tile * 16 + ((lane >> 4) ? 8 : 0);
#pragma unroll
      for (int g = 0; g < 8; ++g) zs[mbase + g][n] = d[g];
    }
  }
  __syncthreads();

  const float p0 = 1.f / (1.f + expf(-zs[lane][0]));
  const float p1 = 1.f / (1.f + expf(-zs[lane][1]));
  const bool mask = tg[lane] > step;
  Dm[lane] = mask ? p0 * 0.2f : 0.f;     // D_SCALE; zero => bit-exact pass-through
  rhom[lane] = mask ? p1 * 0.1f : 0.f;   // RHO_SCALE
}

// ---- stencil: u += D*lap(u) + rho*u*(1-u), periodic; fused output reduction -
__global__ __launch_bounds__(256) void stencil_step(const float* __restrict__ src, long sstride,
                                                    float* __restrict__ dst, long dstride,
                                                    const float* __restrict__ Dm,
                                                    const float* __restrict__ rhom,
                                                    float* __restrict__ partialsOut) {
  __shared__ float tile[(SROWS + 2) * WID];  // 10 rows with halo, 20 KB
  __shared__ float red[256];
  const int b = blockIdx.y;
  const int r0 = blockIdx.x * SROWS;
  const float* sb = src + (long)b * sstride;
  float* db = dst + (long)b * dstride;
  const int tid = threadIdx.x;

  // stage 10 rows (r0-1 .. r0+SROWS) into LDS: 1280 float4, 5 per thread
#pragma unroll
  for (int k = 0; k < 5; ++k) {
    const int idx = tid + k * 256;          // float4 index 0..1279
    const int lr = idx >> 7;                // 128 float4 per row
    const int c4 = idx & 127;
    const int gr = (r0 - 1 + lr + HGT) & (HGT - 1);  // vertical wrap
    lds_copy_b128(sb + (long)gr * WID + c4 * 4, &tile[lr * WID + c4 * 4]);
  }
  lds_copy_wait();

  const float Dv = Dm[b];
  const float Rv = rhom[b];
  float acc = 0.f;
#pragma unroll
  for (int k = 0; k < 16; ++k) {
    const int idx = tid + k * 256;   // 0..4095
    const int lr = idx >> 9;         // 0..7
    const int col = idx & 511;
    const float u = tile[(lr + 1) * WID + col];
    const float lft = tile[(lr + 1) * WID + ((col - 1) & 511)];
    const float rgt = tile[(lr + 1) * WID + ((col + 1) & 511)];
    const float up = tile[lr * WID + col];
    const float dn = tile[(lr + 2) * WID + col];
    const float lap = lft + rgt + up + dn - 4.f * u;
    const float v = u + (Dv * lap + Rv * u * (1.f - u));
    db[(long)(r0 + lr) * WID + col] = v;
    acc += v;
  }

  // fused partial sum of the NEW field (next step's mean)
  red[tid] = acc;
  __syncthreads();
  for (int o = 128; o > 0; o >>= 1) {
    if (tid < o) red[tid] += red[tid + o];
    __syncthreads();
  }
  if (tid == 0) partialsOut[(long)b * NBLK + blockIdx.x] = red[0];
}

extern "C" void kernel_launch(void* const* d_in, const int* in_sizes, int n_in,
                              void* d_out, int out_size, void* d_ws, size_t ws_size,
                              hipStream_t stream) {
  (void)in_sizes; (void)n_in; (void)out_size; (void)ws_size;
  const float* x = (const float*)d_in[0];   // (32,4,512,512)
  const float* Wg = (const float*)d_in[1];  // (4,2)
  const float* bg = (const float*)d_in[2];  // (2,)
  const int* tg = (const int*)d_in[3];      // (32,)
  float* out = (float*)d_out;
  float* ws = (float*)d_ws;

  float* ping      = ws;                                  // 32 MB channel-0 ping buffer
  float* partials  = ping + (size_t)BATCH * PLANE;        // BATCH*NBLK
  float* partials3 = partials + (size_t)BATCH * NBLK;     // BATCH*3*NBLK
  float* chanMeans = partials3 + (size_t)BATCH * 3 * NBLK;// BATCH*CH
  float* Dm        = chanMeans + (size_t)BATCH * CH;      // BATCH
  float* rhom      = Dm + BATCH;                          // BATCH

  // Channels 1..3: copy-through + per-channel means (computed once).
  copy_partial_ch123<<<dim3(NBLK, 3, BATCH), 256, 0, stream>>>(x, out, partials3);
  finalize_ch123<<<1, 96, 0, stream>>>(partials3, chanMeans);

  // Step-0 channel-0 mean partials come from x; later steps get them fused
  // from the previous stencil's output.
  reduce_plane<<<dim3(NBLK, BATCH), 256, 0, stream>>>(x, (long)CH * PLANE, partials);

  for (int s = 0; s < STEPS; ++s) {
    const float* src; long sstr;
    float* dst; long dstr;
    if (s == 0)      { src = x;    sstr = (long)CH * PLANE; }
    else if (s & 1)  { src = ping; sstr = PLANE; }
    else             { src = out;  sstr = (long)CH * PLANE; }
    if (s & 1)       { dst = out;  dstr = (long)CH * PLANE; }
    else             { dst = ping; dstr = PLANE; }

    logit_head<<<1, 32, 0, stream>>>(partials, chanMeans, Wg, bg, tg, s, Dm, rhom);
    stencil_step<<<dim3(HGT / SROWS, BATCH), 256, 0, stream>>>(src, sstr, dst, dstr,
                                                               Dm, rhom, partials);
  }
  // step 7 (odd) writes dst = out channel 0 -> final result in d_out.
}